// Pop2PianoAttention_55336358642501
// MI455X (gfx1250) — compile-verified
//
#include <hip/hip_runtime.h>
#include <hip/hip_bf16.h>
#include <math.h>

typedef __attribute__((ext_vector_type(2))) float v2f;
typedef __attribute__((ext_vector_type(8))) float v8f;

#define B_   2
#define S_   2048
#define D_   1024
#define H_   16
#define DK_  64
#define BIAS_LEN 4095   // (k - q) in [-2047, 2047]

// ---------------------------------------------------------------------------
// CDNA5 async global->LDS copy (16B per lane, tracked by ASYNCcnt).
// Low 32 bits of a generic pointer to __shared__ are the LDS byte address
// (ISA: flat->LDS mapping uses addr[31:0]).
// ---------------------------------------------------------------------------
__device__ __forceinline__ void async_g2l_b128(void* lds_ptr, const void* gptr) {
    unsigned lds = (unsigned)(unsigned long long)lds_ptr;
    asm volatile("global_load_async_to_lds_b128 %0, %1, off"
                 :: "v"(lds), "v"(gptr) : "memory");
}
__device__ __forceinline__ void wait_async0() {
#if __has_builtin(__builtin_amdgcn_s_wait_asynccnt)
    __builtin_amdgcn_s_wait_asynccnt(0);
#else
    asm volatile("s_wait_asynccnt 0x0" ::: "memory");
#endif
}

// ---------------------------------------------------------------------------
// T5 relative-position bias table: biasTab[h][d + 2047], d = k - q
// ---------------------------------------------------------------------------
__global__ void bias_table_kernel(const float* __restrict__ rel_emb,
                                  float* __restrict__ biasTab) {
    int i = blockIdx.x * blockDim.x + threadIdx.x;
    if (i >= H_ * BIAS_LEN) return;
    int h = i / BIAS_LEN;
    int t = i - h * BIAS_LEN;
    int d = t - (S_ - 1);
    int rb = (d > 0) ? 16 : 0;           // bidirectional: num_buckets//2 = 16
    int ad = (d < 0) ? -d : d;
    int bucket;
    if (ad < 8) {                        // max_exact = 8
        bucket = rb + ad;
    } else {
        int lb = 8 + (int)(logf((float)ad / 8.0f) / logf(16.0f) * 8.0f);
        bucket = rb + (lb < 15 ? lb : 15);
    }
    biasTab[i] = rel_emb[bucket * H_ + h];
}

// ---------------------------------------------------------------------------
// C[4096,1024] = X[4096,1024] @ W[1024,1024], fp32 WMMA 16x16x4.
// mode 0: scatter output into [B,H,S,DK] (for Q/K/V)
// mode 1: plain row-major [4096,1024] (final projection)
// WG = 128 threads (4 waves). Tile 64(M) x 64(N), K staged in LDS by 64
// via async global->LDS b128 copies.
// ---------------------------------------------------------------------------
__global__ __launch_bounds__(128) void gemm_f32_wmma(
        const float* __restrict__ X, const float* __restrict__ W,
        float* __restrict__ out, int mode) {
    __shared__ float As[64][68];   // +4 pad: conflict-free column reads
    __shared__ float Bs[64][68];
    const int tid  = threadIdx.x;
    const int wave = tid >> 5;
    const int lane = tid & 31;
    const int half = lane >> 4;    // 0: K={0,1} / M=r ; 1: K={2,3} / M=r+8
    const int lm   = lane & 15;
    const int nb = blockIdx.x * 64;
    const int mb = blockIdx.y * 64;

    v8f acc[4];
    for (int t = 0; t < 4; ++t) acc[t] = {};

    for (int kc = 0; kc < D_; kc += 64) {
        __syncthreads();   // previous tile fully consumed by all waves
        for (int i = 0; i < 8; ++i) {
            int idx = tid + i * 128;          // 0..1023 (64 rows x 16 float4)
            int r = idx >> 4, c4 = (idx & 15) << 2;
            async_g2l_b128(&As[r][c4], X + (size_t)(mb + r) * D_ + kc + c4);
            async_g2l_b128(&Bs[r][c4], W + (size_t)(kc + r) * D_ + nb + c4);
        }
        wait_async0();
        __syncthreads();
        for (int kk = 0; kk < 64; kk += 4) {
            // A-frag (16x4): lane row = lm, cols kk + half*2 + {0,1}
            v2f a = { As[wave * 16 + lm][kk + half * 2],
                      As[wave * 16 + lm][kk + half * 2 + 1] };
            for (int nt = 0; nt < 4; ++nt) {
                // B-frag (4x16): lane col = lm, rows kk + half*2 + {0,1}
                v2f b = { Bs[kk + half * 2][nt * 16 + lm],
                          Bs[kk + half * 2 + 1][nt * 16 + lm] };
                acc[nt] = __builtin_amdgcn_wmma_f32_16x16x4_f32(
                    false, a, false, b, (short)0, acc[nt], false, false);
            }
        }
    }
    // D layout: VGPR r, lane<16 -> M=r ; lane>=16 -> M=r+8 ; N = lm
    for (int nt = 0; nt < 4; ++nt) {
        for (int r = 0; r < 8; ++r) {
            int gm = mb + wave * 16 + r + half * 8;
            int gn = nb + nt * 16 + lm;
            float v = acc[nt][r];
            if (mode == 0) {
                int b = gm >> 11, s = gm & (S_ - 1);
                int h = gn >> 6,  dk = gn & (DK_ - 1);
                out[(((size_t)(b * H_ + h)) * S_ + s) * DK_ + dk] = v;
            } else {
                out[(size_t)gm * D_ + gn] = v;
            }
        }
    }
}

// ---------------------------------------------------------------------------
// Flash attention with T5 relative bias.
// Grid: (S/128, B*H). WG = 256 threads = 8 waves; each wave owns 16 q-rows
// (128-row q-tile halves K/V L2 traffic vs 64). 64-key K/V tiles streamed
// through LDS via async copies; [S,S] scores never materialized.
// ---------------------------------------------------------------------------
__global__ __launch_bounds__(256) void flash_attn_kernel(
        const float* __restrict__ Q, const float* __restrict__ K,
        const float* __restrict__ V, const float* __restrict__ biasTab,
        float* __restrict__ attnOut) {
    __shared__ float Ks[64][68];
    __shared__ float Vs[64][68];
    __shared__ float Ps[8][16][17];   // per-wave P tile (D-layout -> A-layout)
    const int tid  = threadIdx.x;
    const int wave = tid >> 5;
    const int lane = tid & 31;
    const int half = lane >> 4;
    const int lm   = lane & 15;
    const int bh = blockIdx.y;
    const int b  = bh >> 4, h = bh & 15;
    const int q0 = blockIdx.x * 128;

    const float* Qp   = Q + (size_t)bh * S_ * DK_;
    const float* Kp   = K + (size_t)bh * S_ * DK_;
    const float* Vp   = V + (size_t)bh * S_ * DK_;
    const float* btab = biasTab + (size_t)h * BIAS_LEN;

    // Q fragment in registers, A-layout for all 16 k-steps over DK=64
    float qa[16][2];
    {
        const float* qrow = Qp + (size_t)(q0 + wave * 16 + lm) * DK_;
        for (int kk = 0; kk < 16; ++kk) {
            qa[kk][0] = qrow[kk * 4 + half * 2];
            qa[kk][1] = qrow[kk * 4 + half * 2 + 1];
        }
    }

    v8f oacc[4];
    for (int t = 0; t < 4; ++t) oacc[t] = {};
    float m_run[8], l_run[8];            // state for q-row (r + half*8)
    for (int r = 0; r < 8; ++r) { m_run[r] = -INFINITY; l_run[r] = 0.0f; }

    for (int kt = 0; kt < S_; kt += 64) {
        __syncthreads();
        for (int i = 0; i < 4; ++i) {
            int idx = tid + i * 256;          // 0..1023 (64 rows x 16 float4)
            int r = idx >> 4, c4 = (idx & 15) << 2;
            async_g2l_b128(&Ks[r][c4], Kp + (size_t)(kt + r) * DK_ + c4);
            async_g2l_b128(&Vs[r][c4], Vp + (size_t)(kt + r) * DK_ + c4);
        }
        wait_async0();
        __syncthreads();

        for (int j = 0; j < 4; ++j) {            // 16-key sub-tiles
            // S = Q(16xDK) @ K^T(DKx16): 16 WMMA steps of K-dim 4
            v8f sc = {};
            for (int kk = 0; kk < 16; ++kk) {
                v2f a = { qa[kk][0], qa[kk][1] };
                v2f bb = { Ks[j * 16 + lm][kk * 4 + half * 2],
                           Ks[j * 16 + lm][kk * 4 + half * 2 + 1] };
                sc = __builtin_amdgcn_wmma_f32_16x16x4_f32(
                    false, a, false, bb, (short)0, sc, false, false);
            }
            const int kg = kt + j * 16 + lm;     // key index of this lane (N)
            // online softmax, rows live across the 16 lanes of a half-wave
            for (int r = 0; r < 8; ++r) {
                int qg = q0 + wave * 16 + r + half * 8;
                float s = sc[r] + btab[kg - qg + (S_ - 1)];
                float mx = s;
                for (int m = 1; m < 16; m <<= 1)
                    mx = fmaxf(mx, __shfl_xor(mx, m, 16));
                float mnew  = fmaxf(m_run[r], mx);
                float scale = expf(m_run[r] - mnew);
                float pv    = expf(s - mnew);
                float rs = pv;
                for (int m = 1; m < 16; m <<= 1)
                    rs += __shfl_xor(rs, m, 16);
                l_run[r] = l_run[r] * scale + rs;
                m_run[r] = mnew;
                for (int nt = 0; nt < 4; ++nt) oacc[nt][r] *= scale;
                Ps[wave][r + half * 8][lm] = pv;  // D-layout -> LDS
            }
            // O += P(16x16) @ V(16x64); P re-read in A-layout (same-wave LDS,
            // DS ops are in-order within a wave)
            for (int kk2 = 0; kk2 < 16; kk2 += 4) {
                v2f a = { Ps[wave][lm][kk2 + half * 2],
                          Ps[wave][lm][kk2 + half * 2 + 1] };
                for (int nt = 0; nt < 4; ++nt) {
                    v2f bb = { Vs[j * 16 + kk2 + half * 2][nt * 16 + lm],
                               Vs[j * 16 + kk2 + half * 2 + 1][nt * 16 + lm] };
                    oacc[nt] = __builtin_amdgcn_wmma_f32_16x16x4_f32(
                        false, a, false, bb, (short)0, oacc[nt], false, false);
                }
            }
        }
    }
    // normalize and write [B*S, H*DK] for the final projection
    for (int nt = 0; nt < 4; ++nt) {
        for (int r = 0; r < 8; ++r) {
            int qg  = q0 + wave * 16 + r + half * 8;
            int row = b * S_ + qg;
            int col = h * DK_ + nt * 16 + lm;
            attnOut[(size_t)row * D_ + col] = oacc[nt][r] / l_run[r];
        }
    }
}

// ---------------------------------------------------------------------------
extern "C" void kernel_launch(void* const* d_in, const int* in_sizes, int n_in,
                              void* d_out, int out_size, void* d_ws, size_t ws_size,
                              hipStream_t stream) {
    const float* hs  = (const float*)d_in[0];
    const float* wq  = (const float*)d_in[1];
    const float* wk  = (const float*)d_in[2];
    const float* wv  = (const float*)d_in[3];
    const float* wo  = (const float*)d_in[4];
    const float* rel = (const float*)d_in[5];

    float* ws = (float*)d_ws;
    const size_t qkv = (size_t)B_ * H_ * S_ * DK_;   // 4Mi floats each
    float* Qb = ws;
    float* Kb = Qb + qkv;
    float* Vb = Kb + qkv;
    float* Ab = Vb + qkv;                            // [B*S, H*DK]
    float* Bt = Ab + (size_t)B_ * S_ * H_ * DK_;     // bias table

    bias_table_kernel<<<(H_ * BIAS_LEN + 255) / 256, 256, 0, stream>>>(rel, Bt);

    dim3 gg(D_ / 64, (B_ * S_) / 64), blk(128);
    gemm_f32_wmma<<<gg, blk, 0, stream>>>(hs, wq, Qb, 0);
    gemm_f32_wmma<<<gg, blk, 0, stream>>>(hs, wk, Kb, 0);
    gemm_f32_wmma<<<gg, blk, 0, stream>>>(hs, wv, Vb, 0);

    flash_attn_kernel<<<dim3(S_ / 128, B_ * H_), dim3(256), 0, stream>>>(Qb, Kb, Vb, Bt, Ab);

    gemm_f32_wmma<<<gg, blk, 0, stream>>>(Ab, wo, (float*)d_out, 1);
}